// FrequencyAdaptiveNormSimple_14353780704084
// MI455X (gfx1250) — compile-verified
//
#include <hip/hip_runtime.h>
#include <cstdint>

typedef __attribute__((ext_vector_type(2))) float v2f;
typedef __attribute__((ext_vector_type(4))) float v4f;
typedef __attribute__((ext_vector_type(8))) float v8f;

#define B_   32
#define T_   2048
#define H_   512
#define TBLK 128           // time extent per workgroup
#define HALO 20            // max window - 0 (front halo rows)
#define ROWS (TBLK + HALO) // 148 LDS rows
#define HT   16            // feature tile width
#define EPS  1e-4f

// Per-window contribution for one output element.
// Edge fix: for t < w-1 the reference replicates the stat of window [0, w-1];
// those rows live at LDS rows HALO..HALO+w-1 (only reachable when tb == 0,
// which is exactly when tg < w-1 can occur).
__device__ __forceinline__ float win_contrib(const float* tile, float s, float s2,
                                             int w, float invw, float invwm1,
                                             float alw, int tg, float xv, int n) {
  if (tg < w - 1) {
    s = 0.f; s2 = 0.f;
    for (int r = 0; r < w; ++r) {
      float v = tile[(HALO + r) * HT + n];
      s += v; s2 += v * v;
    }
  }
  float mu  = s * invw;
  float var = (s2 - s * mu) * invwm1;   // unbiased
  var = fmaxf(var, 0.f);
  float sd = fmaxf(__builtin_sqrtf(var), EPS);
  float z  = (xv - mu) / sd;            // finite: sd >= EPS
  return z * alw;
}

__global__ __launch_bounds__(256, 1)
void fan_kernel(const float* __restrict__ xg,
                const float* __restrict__ alpha,
                float* __restrict__ outg) {
  // Single shared object -> assumed at LDS byte offset 0 (used by async-load VDST).
  __shared__ float tile[ROWS * HT];

  const int tid = threadIdx.x;
  const int h0  = blockIdx.x * HT;
  const int tb  = blockIdx.y * TBLK;
  const int b   = blockIdx.z;

  // ---------- Stage x[tb-20 .. tb+127][h0..h0+15] into LDS via async tensor path ----------
  // 148 rows * 64B = 592 16-byte segments; 256 lanes -> 3 sweeps.
  for (int it = 0; it < 3; ++it) {
    int seg = tid + it * 256;
    if (seg < ROWS * 4) {
      int row  = seg >> 2;
      int part = seg & 3;
      int tg   = tb - HALO + row;
      if (tg >= 0) {
        uint64_t ga = (uint64_t)(uintptr_t)xg +
                      (((uint64_t)b * T_ + (uint64_t)tg) * H_ + (uint64_t)(h0 + part * 4)) * 4ull;
        unsigned ldsOff = (unsigned)((row * HT + part * 4) * sizeof(float));
        asm volatile("global_load_async_to_lds_b128 %0, %1, off"
                     :: "v"(ldsOff), "v"(ga) : "memory");
      } else {
        v4f z = {0.f, 0.f, 0.f, 0.f};
        *(v4f*)(&tile[row * HT + part * 4]) = z;  // halo rows before t=0 contribute 0
      }
    }
  }
  asm volatile("s_wait_asynccnt 0" ::: "memory");
  __syncthreads();

  // ---------- Each wave computes one 16(T) x 16(H) output tile with f32 WMMA ----------
  const int lane  = tid & 31;
  const int wid   = tid >> 5;       // wave id = t-subtile index 0..7
  const int n     = lane & 15;      // N for B/C/D layouts, M for A layout
  const int kh    = lane >> 4;      // lane half: selects K pair (A/B) and M half (C/D)
  const int rbase = wid * 16;       // local row of region start (tile t0 - 20)

  v8f S5 = {}, S10 = {}, S20 = {};  // window sums
  v8f Q5 = {}, Q10 = {}, Q20 = {};  // window sums of squares

#pragma unroll
  for (int c = 0; c < 9; ++c) {     // nine K=4 chunks covering 36 halo-region rows
    const int rowx = rbase + c * 4 + kh * 2;
    v2f bx, bq;
    // B 4x16 layout: vgpr j holds row K = j + 2*kh, striped over lanes 0..15 (N)
    bx.x = tile[(rowx)     * HT + n];
    bx.y = tile[(rowx + 1) * HT + n];
    bq.x = bx.x * bx.x;
    bq.y = bx.y * bx.y;
    // A 16x4 layout: lane m holds K = 2*kh + j in vgpr j. Region row r = 4c + K.
    // Window w covers r - m in [21-w, 20].
    const int d0 = c * 4 + kh * 2 - n;  // (n == m bits for A lanes)
    const int d1 = d0 + 1;
    v2f a;
    a.x = (d0 >= 1 && d0 <= 20) ? 1.f : 0.f;   // w = 20
    a.y = (d1 >= 1 && d1 <= 20) ? 1.f : 0.f;
    S20 = __builtin_amdgcn_wmma_f32_16x16x4_f32(false, a, false, bx, (short)0, S20, false, false);
    Q20 = __builtin_amdgcn_wmma_f32_16x16x4_f32(false, a, false, bq, (short)0, Q20, false, false);
    if (c >= 2) {                               // w = 10: r-m in [11,20] -> chunks 2..8
      a.x = (d0 >= 11 && d0 <= 20) ? 1.f : 0.f;
      a.y = (d1 >= 11 && d1 <= 20) ? 1.f : 0.f;
      S10 = __builtin_amdgcn_wmma_f32_16x16x4_f32(false, a, false, bx, (short)0, S10, false, false);
      Q10 = __builtin_amdgcn_wmma_f32_16x16x4_f32(false, a, false, bq, (short)0, Q10, false, false);
    }
    if (c >= 4) {                               // w = 5: r-m in [16,20] -> chunks 4..8
      a.x = (d0 >= 16 && d0 <= 20) ? 1.f : 0.f;
      a.y = (d1 >= 16 && d1 <= 20) ? 1.f : 0.f;
      S5 = __builtin_amdgcn_wmma_f32_16x16x4_f32(false, a, false, bx, (short)0, S5, false, false);
      Q5 = __builtin_amdgcn_wmma_f32_16x16x4_f32(false, a, false, bq, (short)0, Q5, false, false);
    }
  }

  // ---------- Per-feature softmax blend weights (WINDOWS = (5, 10, 20)) ----------
  const int h = h0 + n;
  float a5v  = alpha[h * 3 + 0];
  float a10v = alpha[h * 3 + 1];
  float a20v = alpha[h * 3 + 2];
  float mx  = fmaxf(a5v, fmaxf(a10v, a20v));
  float e5  = __expf(a5v - mx);
  float e10 = __expf(a10v - mx);
  float e20 = __expf(a20v - mx);
  float inv = 1.f / (e5 + e10 + e20);
  float al5 = e5 * inv, al10 = e10 * inv, al20 = e20 * inv;

  // ---------- Epilogue: C/D layout vgpr i -> M = i + 8*kh, N = n ----------
#pragma unroll
  for (int i = 0; i < 8; ++i) {
    const int m  = i + 8 * kh;
    const int tg = tb + rbase + m;
    const float xv = tile[(rbase + HALO + m) * HT + n];
    float acc = 0.f;
    acc += win_contrib(tile, S5[i],  Q5[i],   5, 0.2f,  0.25f,       al5,  tg, xv, n);
    acc += win_contrib(tile, S10[i], Q10[i], 10, 0.1f,  1.f / 9.f,   al10, tg, xv, n);
    acc += win_contrib(tile, S20[i], Q20[i], 20, 0.05f, 1.f / 19.f,  al20, tg, xv, n);
    outg[((uint64_t)b * T_ + (uint64_t)tg) * H_ + (uint64_t)h] = acc;
  }
}

extern "C" void kernel_launch(void* const* d_in, const int* in_sizes, int n_in,
                              void* d_out, int out_size, void* d_ws, size_t ws_size,
                              hipStream_t stream) {
  (void)in_sizes; (void)n_in; (void)out_size; (void)d_ws; (void)ws_size;
  const float* x     = (const float*)d_in[0];
  const float* alpha = (const float*)d_in[1];
  float* out         = (float*)d_out;
  dim3 grid(H_ / HT, T_ / TBLK, B_);  // (32, 16, 32)
  fan_kernel<<<grid, dim3(256), 0, stream>>>(x, alpha, out);
}